// TRecPolicy_6811818131652
// MI455X (gfx1250) — compile-verified
//
#include <hip/hip_runtime.h>

// MI455X / gfx1250, wave32. 16 batch rows per wave; all matvecs on the
// matrix pipe via V_WMMA_F32_16X16X4_F32 (fp32-exact, K == H == 4).

typedef __attribute__((ext_vector_type(2))) float v2f;
typedef __attribute__((ext_vector_type(8))) float v8f;

#define WAVES_PER_BLOCK 8
#define TILE_M 16
#define LDS_PER_WAVE (288 + 64)   // 16x18 x-tile + 16x4 h transpose scratch

__device__ __forceinline__ float fast_tanhf(float x) {
#if __has_builtin(__builtin_amdgcn_tanhf)
  return __builtin_amdgcn_tanhf(x);          // v_tanh_f32 (gfx1250)
#else
  float e = __builtin_amdgcn_exp2f(x * -2.885390081777927f);  // exp(-2x)
  return (1.0f - e) * __builtin_amdgcn_rcpf(1.0f + e);
#endif
}

__device__ __forceinline__ float fast_sigmoidf(float x) {
  return 0.5f + 0.5f * fast_tanhf(0.5f * x);
}

__device__ __forceinline__ v8f splat8(float v) {
  v8f r;
#pragma unroll
  for (int i = 0; i < 8; ++i) r[i] = v;
  return r;
}

__device__ __forceinline__ v8f wmma4(v2f a, v2f b, v8f c) {
  // D(16x16) = A(16x4) * B(4x16) + C, fp32
  return __builtin_amdgcn_wmma_f32_16x16x4_f32(
      /*neg_a=*/false, a, /*neg_b=*/false, b,
      /*c_mod=*/(short)0, c, /*reuse_a=*/false, /*reuse_b=*/false);
}

// One GRU elementwise update given the two gate tiles (D layout).
// hD holds old h at columns 8..11 (rows v+8*hi); produces new hD and,
// via LDS transpose, the A-layout hA for the next WMMA. Leaves hswap
// holding hnew[m][k] row-major for head computations.
__device__ __forceinline__ void gru_step(const v8f& gi, const v8f& gh,
                                         v8f& hD, v2f& hA,
                                         float* hswap, int lane)
{
  const int n  = lane & 15;
  const int hi = lane >> 4;
  float sg[8];
#pragma unroll
  for (int v = 0; v < 8; ++v) sg[v] = fast_sigmoidf(gi[v] + gh[v]); // r (cols0-3), z (cols4-7)
  const int src_r = (lane + 24) & 31;  // lane - 8: r column k -> column 8+k
  const int src_z = (lane + 28) & 31;  // lane - 4: z column 4+k -> column 8+k
  v8f hn;
#pragma unroll
  for (int v = 0; v < 8; ++v) {
    float r  = __shfl(sg[v], src_r, 32);
    float z  = __shfl(sg[v], src_z, 32);
    float nn = fast_tanhf(gi[v] + r * gh[v]);    // meaningful at cols 8..11
    hn[v] = (1.0f - z) * nn + z * hD[v];
  }
  hD = hn;
  __builtin_amdgcn_wave_barrier();
  if (n >= 8 && n < 12) {                        // scatter hnew -> LDS as h[m][k]
    const int kk = n - 8;
#pragma unroll
    for (int v = 0; v < 8; ++v) hswap[(v + 8 * hi) * 4 + kk] = hn[v];
  }
  __builtin_amdgcn_wave_barrier();
  hA.x = hswap[n * 4 + 2 * hi + 0];              // gather A layout (m=n, K=2*hi{,+1})
  hA.y = hswap[n * 4 + 2 * hi + 1];
  __builtin_amdgcn_wave_barrier();
}

__global__ void __launch_bounds__(256)
trec_policy_kernel(const float* __restrict__ x,
                   const float* __restrict__ w_ih_up, const float* __restrict__ w_hh_up,
                   const float* __restrict__ b_ih_up, const float* __restrict__ b_hh_up,
                   const float* __restrict__ w1, const float* __restrict__ b1,
                   const float* __restrict__ w2, const float* __restrict__ b2,
                   const float* __restrict__ w_ih_dn, const float* __restrict__ w_hh_dn,
                   const float* __restrict__ b_ih_dn, const float* __restrict__ b_hh_dn,
                   const float* __restrict__ w_out, const float* __restrict__ b_out,
                   float* __restrict__ out, int Btot)
{
  __shared__ float lds[WAVES_PER_BLOCK * LDS_PER_WAVE];
  const int lane = threadIdx.x & 31;
  const int wave = threadIdx.x >> 5;
  const int n    = lane & 15;            // column (B/C/D) or row m (A)
  const int hi   = lane >> 4;
  const int k0   = 2 * hi;               // A/B K-pair base for this half-wave
  float* xs    = &lds[wave * LDS_PER_WAVE];
  float* hswap = xs + 288;

  const long tile = (long)blockIdx.x * WAVES_PER_BLOCK + wave;
  const long base = tile * TILE_M;

  // ---- stage x tile (16 rows x 18 f32) through LDS, coalesced 128B loads ----
  const long xlim = (long)Btot * 18;
#pragma unroll
  for (int t = 0; t < 9; ++t) {
    const int idx = lane + 32 * t;
    const long g  = base * 18 + idx;
    xs[idx] = (g < xlim) ? x[g] : 0.0f;
  }
  __builtin_amdgcn_wave_barrier();

  // ---- wave-constant B matrices (4x16), per-lane entries B[k][n] = W[n][k] ----
  v2f Bihup, Bhhup, Bw1lo, Bw1hi, Bw2, Bihdn, Bhhdn;
  Bihup.x = (hi == 0 && n < 12) ? w_ih_up[n * 2 + 0] : 0.0f;   // K rows 2,3 are pad
  Bihup.y = (hi == 0 && n < 12) ? w_ih_up[n * 2 + 1] : 0.0f;
  Bhhup.x = (n < 12) ? w_hh_up[n * 4 + k0 + 0] : 0.0f;
  Bhhup.y = (n < 12) ? w_hh_up[n * 4 + k0 + 1] : 0.0f;
  Bw1lo.x = (n < 4) ? w1[n * 8 + k0 + 0] : 0.0f;               // in-dims 0..3 (obs)
  Bw1lo.y = (n < 4) ? w1[n * 8 + k0 + 1] : 0.0f;
  Bw1hi.x = (n < 4) ? w1[n * 8 + 4 + k0 + 0] : 0.0f;           // in-dims 4..7 (h)
  Bw1hi.y = (n < 4) ? w1[n * 8 + 4 + k0 + 1] : 0.0f;
  Bw2.x   = (n < 4) ? w2[n * 4 + k0 + 0] : 0.0f;
  Bw2.y   = (n < 4) ? w2[n * 4 + k0 + 1] : 0.0f;
  Bihdn.x = (n < 12) ? w_ih_dn[n * 4 + k0 + 0] : 0.0f;
  Bihdn.y = (n < 12) ? w_ih_dn[n * 4 + k0 + 1] : 0.0f;
  Bhhdn.x = (n < 12) ? w_hh_dn[n * 4 + k0 + 0] : 0.0f;
  Bhhdn.y = (n < 12) ? w_hh_dn[n * 4 + k0 + 1] : 0.0f;

  const float bb_ihup = (n < 12) ? b_ih_up[n] : 0.0f;
  const float bb_hhup = (n < 12) ? b_hh_up[n] : 0.0f;
  const float bb1     = (n < 4)  ? b1[n]      : 0.0f;
  const float bb2     = (n < 4)  ? b2[n]      : 0.0f;
  const float bb_ihdn = (n < 12) ? b_ih_dn[n] : 0.0f;
  const float bb_hhdn = (n < 12) ? b_hh_dn[n] : 0.0f;
  const float wo0 = w_out[0], wo1 = w_out[1], wo2 = w_out[2],
              wo3 = w_out[3], wo4 = w_out[4], bo = b_out[0];

  // ---- up GRU: i = 6..0, H=4, input (j[i], jd[i]) ----
  v2f hA; hA.x = 0.0f; hA.y = 0.0f;
  v8f hD = splat8(0.0f);
  v2f hist[7];
#pragma unroll
  for (int u = 0; u < 7; ++u) {
    const int jj = 6 - u;
    v2f axi;                                    // A = [j, jd, 0, 0] (16x4)
    axi.x = (hi == 0) ? xs[n * 18 + 4  + jj] : 0.0f;
    axi.y = (hi == 0) ? xs[n * 18 + 11 + jj] : 0.0f;
    v8f gi = wmma4(axi, Bihup, splat8(bb_ihup));
    v8f gh = wmma4(hA,  Bhhup, splat8(bb_hhup));
    gru_step(gi, gh, hD, hA, hswap, lane);
    hist[u] = hA;                               // h_up[u] in A layout
  }

  // ---- mid MLP: tanh([obs,h] @ w1^T + b1) -> tanh(@ w2^T + b2) ----
  v2f aobs;                                     // obs = x[:,0:4]
  aobs.x = xs[n * 18 + k0 + 0];
  aobs.y = xs[n * 18 + k0 + 1];
  v8f t1 = wmma4(hA,   Bw1hi, splat8(bb1));     // K split 4+4, chained via C
  t1     = wmma4(aobs, Bw1lo, t1);
  v8f h2;
#pragma unroll
  for (int v = 0; v < 8; ++v) h2[v] = fast_tanhf(t1[v]);   // cols 0..3
  __builtin_amdgcn_wave_barrier();
  if (n < 4) {
#pragma unroll
    for (int v = 0; v < 8; ++v) hswap[(v + 8 * hi) * 4 + n] = h2[v];
  }
  __builtin_amdgcn_wave_barrier();
  v2f a2; a2.x = hswap[n * 4 + k0]; a2.y = hswap[n * 4 + k0 + 1];
  __builtin_amdgcn_wave_barrier();
  v8f t2 = wmma4(a2, Bw2, splat8(bb2));
  v8f h3;
#pragma unroll
  for (int v = 0; v < 8; ++v) h3[v] = fast_tanhf(t2[v]);   // cols 0..3

  // seed down-state: hD wants h at cols 8..11; hA wants A layout
  const int src_h = (lane + 24) & 31;           // lane - 8
#pragma unroll
  for (int v = 0; v < 8; ++v) hD[v] = __shfl(h3[v], src_h, 32);
  __builtin_amdgcn_wave_barrier();
  if (n < 4) {
#pragma unroll
    for (int v = 0; v < 8; ++v) hswap[(v + 8 * hi) * 4 + n] = h3[v];
  }
  __builtin_amdgcn_wave_barrier();
  hA.x = hswap[n * 4 + k0]; hA.y = hswap[n * 4 + k0 + 1];
  __builtin_amdgcn_wave_barrier();

  // ---- down GRU + per-step linear head ----
  float acts[7];
#pragma unroll
  for (int i = 0; i < 7; ++i) {
    v8f gi = wmma4(hist[i], Bihdn, splat8(bb_ihdn));
    v8f gh = wmma4(hA,      Bhhdn, splat8(bb_hhdn));
    gru_step(gi, gh, hD, hA, hswap, lane);
    float a = 0.0f;
    if (hi == 0) {                              // hswap holds hnew[m][k]
      a = bo + hswap[n * 4 + 0] * wo0 + hswap[n * 4 + 1] * wo1
             + hswap[n * 4 + 2] * wo2 + hswap[n * 4 + 3] * wo3
             + xs[n * 18 + 4 + i] * wo4;        // concat with j[:, i]
    }
    acts[i] = a;
  }

  if (hi == 0) {
    const long row = base + n;
    if (row < (long)Btot) {
#pragma unroll
      for (int i = 0; i < 7; ++i) out[row * 7 + i] = acts[i];
    }
  }
}

extern "C" void kernel_launch(void* const* d_in, const int* in_sizes, int n_in,
                              void* d_out, int out_size, void* d_ws, size_t ws_size,
                              hipStream_t stream) {
  const float* x       = (const float*)d_in[0];
  const float* w_ih_up = (const float*)d_in[1];
  const float* w_hh_up = (const float*)d_in[2];
  const float* b_ih_up = (const float*)d_in[3];
  const float* b_hh_up = (const float*)d_in[4];
  const float* w1      = (const float*)d_in[5];
  const float* b1      = (const float*)d_in[6];
  const float* w2      = (const float*)d_in[7];
  const float* b2      = (const float*)d_in[8];
  const float* w_ih_dn = (const float*)d_in[9];
  const float* w_hh_dn = (const float*)d_in[10];
  const float* b_ih_dn = (const float*)d_in[11];
  const float* b_hh_dn = (const float*)d_in[12];
  const float* w_out   = (const float*)d_in[13];
  const float* b_out   = (const float*)d_in[14];
  float* out = (float*)d_out;

  const int Btot   = in_sizes[0] / 18;
  const int tiles  = (Btot + TILE_M - 1) / TILE_M;
  const int blocks = (tiles + WAVES_PER_BLOCK - 1) / WAVES_PER_BLOCK;

  trec_policy_kernel<<<dim3(blocks), dim3(256), 0, stream>>>(
      x, w_ih_up, w_hh_up, b_ih_up, b_hh_up, w1, b1, w2, b2,
      w_ih_dn, w_hh_dn, b_ih_dn, b_hh_dn, w_out, b_out, out, Btot);
}